// StackAugmentedVAE_3753801416872
// MI455X (gfx1250) — compile-verified
//
#include <hip/hip_runtime.h>
#include <hip/hip_bf16.h>
#include <math.h>

// ---------------------------------------------------------------------------
// StackAugmentedVAE single step on MI455X (gfx1250, wave32).
// HBM-bandwidth-bound GEMV chain (~213 MB fp32 weights, ~9us floor @23.3TB/s).
// GEMVs use V_WMMA_F32_16X16X4_F32 with 4 interleaved accumulators,
// K-split partial sums for occupancy, LDS-staged x vector, and prefetch.
// ---------------------------------------------------------------------------

typedef __attribute__((ext_vector_type(2))) float v2f;
typedef __attribute__((ext_vector_type(8))) float v8f;

#define VOCAB 100
#define H     2048
#define NL    2
#define SW    256
#define SD    200
#define RIN   (H + SW)   // 2304

// ---------------------------------------------------------------------------
// GEMV partial: y[blockIdx.y*R + r] = dot(W[r, kc..kc+KC), x[kc..kc+KC))
// (bias added by split 0 only, when provided).
//
// Wave handles 16 consecutive rows. A operand (16x4 f32 tile): lanes 0-15
// carry (row, K=k..k+1), lanes 16-31 carry (row-16, K=k+2..k+3) -> v2f/lane.
// B operand: x[k..k+3] replicated across all 16 columns, read from LDS.
// Every D column holds the 16 dot products; lanes 0/16 store M=0..7 / 8..15.
// ---------------------------------------------------------------------------
template<int K_TOTAL, int NSPLIT>
__global__ __launch_bounds__(256)
void gemv_wmma_kernel(const float* __restrict__ W,
                      const float* __restrict__ x,
                      const float* __restrict__ bias,   // may be nullptr
                      float* __restrict__ y,            // NSPLIT partial buffers
                      int R)
{
    constexpr int KC = K_TOTAL / NSPLIT;                // multiple of 16
    __shared__ float xs[KC];
    const int kStart = blockIdx.y * KC;

    // Stage this block's x chunk into LDS (coalesced float4).
    for (int i = threadIdx.x; i < KC / 4; i += blockDim.x)
        ((float4*)xs)[i] = ((const float4*)(x + kStart))[i];
    __syncthreads();

    const int lane    = threadIdx.x & 31;
    const int wave    = threadIdx.x >> 5;
    const int rowBase = (blockIdx.x * (blockDim.x >> 5) + wave) * 16;
    if (rowBase >= R) return;                           // uniform per-wave exit

    const int m    = lane & 15;
    const int koff = (lane >> 4) * 2;                   // 0 | 2 per half-wave
    int row = rowBase + m;
    if (row > R - 1) row = R - 1;                       // clamp ragged tail loads

    const float* wp = W + (size_t)row * K_TOTAL + kStart + koff;
    const float* xp = xs + koff;

    v8f a0 = {}, a1 = {}, a2 = {}, a3 = {};
    #pragma unroll 2
    for (int k = 0; k < KC; k += 16) {
        // run ~2KB ahead of each of the 16 row streams
        __builtin_prefetch((const void*)(wp + k + 512), 0, 1);
        v2f w0 = *(const v2f*)(wp + k);
        v2f w1 = *(const v2f*)(wp + k + 4);
        v2f w2 = *(const v2f*)(wp + k + 8);
        v2f w3 = *(const v2f*)(wp + k + 12);
        v2f b0 = *(const v2f*)(xp + k);
        v2f b1 = *(const v2f*)(xp + k + 4);
        v2f b2 = *(const v2f*)(xp + k + 8);
        v2f b3 = *(const v2f*)(xp + k + 12);
        a0 = __builtin_amdgcn_wmma_f32_16x16x4_f32(false, w0, false, b0, (short)0, a0, false, false);
        a1 = __builtin_amdgcn_wmma_f32_16x16x4_f32(false, w1, false, b1, (short)0, a1, false, false);
        a2 = __builtin_amdgcn_wmma_f32_16x16x4_f32(false, w2, false, b2, (short)0, a2, false, false);
        a3 = __builtin_amdgcn_wmma_f32_16x16x4_f32(false, w3, false, b3, (short)0, a3, false, false);
    }
    v8f acc = (a0 + a1) + (a2 + a3);

    if (lane == 0 || lane == 16) {
        float* yout = y + (size_t)blockIdx.y * R;
        const int mBase = rowBase + ((lane >> 4) * 8);
        #pragma unroll
        for (int i = 0; i < 8; ++i) {
            const int r = mBase + i;
            if (r < R) {
                float v = acc[i];
                if (bias && blockIdx.y == 0) v += bias[r];
                yout[r] = v;
            }
        }
    }
}

// ---------------------------------------------------------------------------
// Controls: softmax(Wc @ h2s + bc).  h2s == hidden flattened (4096).
// ---------------------------------------------------------------------------
__global__ __launch_bounds__(256)
void controls_kernel(const float* __restrict__ hidden,
                     const float* __restrict__ Wc,
                     const float* __restrict__ bc,
                     float* __restrict__ ctrl)          // [push, pop, noop]
{
    __shared__ float red[3][256];
    const int t = threadIdx.x;
    float s0 = 0.f, s1 = 0.f, s2 = 0.f;
    for (int k = t; k < NL * H; k += 256) {
        const float h = hidden[k];
        s0 += Wc[k] * h;
        s1 += Wc[NL * H + k] * h;
        s2 += Wc[2 * NL * H + k] * h;
    }
    red[0][t] = s0; red[1][t] = s1; red[2][t] = s2;
    __syncthreads();
    for (int s = 128; s > 0; s >>= 1) {
        if (t < s) {
            red[0][t] += red[0][t + s];
            red[1][t] += red[1][t + s];
            red[2][t] += red[2][t + s];
        }
        __syncthreads();
    }
    if (t == 0) {
        const float c0 = red[0][0] + bc[0];
        const float c1 = red[1][0] + bc[1];
        const float c2 = red[2][0] + bc[2];
        const float mx = fmaxf(c0, fmaxf(c1, c2));
        const float e0 = expf(c0 - mx), e1 = expf(c1 - mx), e2 = expf(c2 - mx);
        const float inv = 1.f / (e0 + e1 + e2);
        ctrl[0] = e0 * inv;   // a_push
        ctrl[1] = e1 * inv;   // a_pop
        ctrl[2] = e2 * inv;   // a_noop
    }
}

// ---------------------------------------------------------------------------
// x[0..H-1] = emb[inp]
// ---------------------------------------------------------------------------
__global__ __launch_bounds__(256)
void embed_kernel(const int* __restrict__ inp,
                  const float* __restrict__ emb,
                  float* __restrict__ xvec)
{
    const int i = blockIdx.x * blockDim.x + threadIdx.x;
    if (i < H) xvec[i] = emb[(size_t)inp[0] * H + i];
}

// ---------------------------------------------------------------------------
// Stack blend; sums the 8 K-split partials of s_raw for the tanh input.
// blockIdx.x = depth d, threadIdx.x = w. Row 0 also feeds x[H..H+SW-1].
// ---------------------------------------------------------------------------
__global__ __launch_bounds__(SW)
void stack_kernel(const float* __restrict__ stack,    // SD x SW (input)
                  const float* __restrict__ s_raw,    // 8 x SW partials
                  const float* __restrict__ bs,
                  const float* __restrict__ ctrl,
                  float* __restrict__ new_stack,      // SD x SW (output)
                  float* __restrict__ xvec)           // RIN
{
    const int d = blockIdx.x;
    const int w = threadIdx.x;
    const float a_push = ctrl[0], a_pop = ctrl[1], a_noop = ctrl[2];

    float up;
    if (d == 0) {
        float s = bs[w];
        #pragma unroll
        for (int p = 0; p < 8; ++p) s += s_raw[p * SW + w];
        up = tanhf(s);
    } else {
        up = stack[(d - 1) * SW + w];
    }
    const float down = (d == SD - 1) ? 0.f : stack[(d + 1) * SW + w];
    const float ns   = a_noop * stack[d * SW + w] + a_push * up + a_pop * down;

    new_stack[d * SW + w] = ns;
    if (d == 0) xvec[H + w] = ns;
}

// ---------------------------------------------------------------------------
// GRU gate combine; sums 4 K-split partials of gi and gh.
// ---------------------------------------------------------------------------
__device__ __forceinline__ float sigmoidf(float v) { return 1.f / (1.f + expf(-v)); }

__global__ __launch_bounds__(256)
void gru_gate_kernel(const float* __restrict__ gi,    // 4 x 3H partials
                     const float* __restrict__ gh,    // 4 x 3H partials
                     const float* __restrict__ bih,
                     const float* __restrict__ bhh,
                     const float* __restrict__ hprev, // H
                     float* __restrict__ hout,        // H (d_out region)
                     float* __restrict__ hws)         // H (ws copy, next GEMV)
{
    const int j = blockIdx.x * blockDim.x + threadIdx.x;
    if (j >= H) return;
    float ir = bih[j],         hr = bhh[j];
    float iz = bih[H + j],     hz = bhh[H + j];
    float in = bih[2 * H + j], hn = bhh[2 * H + j];
    #pragma unroll
    for (int p = 0; p < 4; ++p) {
        const float* gip = gi + p * 3 * H;
        const float* ghp = gh + p * 3 * H;
        ir += gip[j];         hr += ghp[j];
        iz += gip[H + j];     hz += ghp[H + j];
        in += gip[2 * H + j]; hn += ghp[2 * H + j];
    }
    const float r = sigmoidf(ir + hr);
    const float z = sigmoidf(iz + hz);
    const float n = tanhf(in + r * hn);
    const float h = (1.f - z) * n + z * hprev[j];
    hout[j] = h;
    hws[j]  = h;
}

// ---------------------------------------------------------------------------
// Sum the 4 logits partials (bias already folded into partial 0).
// ---------------------------------------------------------------------------
__global__ __launch_bounds__(128)
void logits_final_kernel(const float* __restrict__ part, float* __restrict__ out)
{
    const int j = threadIdx.x;
    if (j < VOCAB)
        out[j] = part[j] + part[VOCAB + j] + part[2 * VOCAB + j] + part[3 * VOCAB + j];
}

// ---------------------------------------------------------------------------
// Host-side launch sequence (stream-ordered dependencies).
// ---------------------------------------------------------------------------
extern "C" void kernel_launch(void* const* d_in, const int* in_sizes, int n_in,
                              void* d_out, int out_size, void* d_ws, size_t ws_size,
                              hipStream_t stream)
{
    (void)in_sizes; (void)n_in; (void)out_size; (void)ws_size;

    const int*   inp    = (const int*)  d_in[0];
    const float* hidden = (const float*)d_in[1];   // (2,1,2048) flat == h2s
    const float* stack  = (const float*)d_in[2];
    const float* emb    = (const float*)d_in[3];
    const float* Wc     = (const float*)d_in[4];
    const float* bc     = (const float*)d_in[5];
    const float* Ws     = (const float*)d_in[6];   // (256,4096)
    const float* bs     = (const float*)d_in[7];
    const float* Wih0   = (const float*)d_in[8];   // (6144,2304)
    const float* Whh0   = (const float*)d_in[9];   // (6144,2048)
    const float* bih0   = (const float*)d_in[10];
    const float* bhh0   = (const float*)d_in[11];
    const float* Wih1   = (const float*)d_in[12];  // (6144,2048)
    const float* Whh1   = (const float*)d_in[13];
    const float* bih1   = (const float*)d_in[14];
    const float* bhh1   = (const float*)d_in[15];
    const float* Wd     = (const float*)d_in[16];  // (100,2048)
    const float* bd     = (const float*)d_in[17];

    // Output: logits(100) | next_hidden(2*2048) | new_stack(200*256)
    float* out_logits = (float*)d_out;
    float* out_hidden = out_logits + VOCAB;
    float* out_stack  = out_hidden + NL * H;

    // Workspace (floats; all region starts 16-float aligned)
    float* ws    = (float*)d_ws;
    float* ctrl  = ws;              // 3 (+pad)
    float* s_raw = ws + 16;         // 8 x 256   = 2048
    float* x0    = ws + 2064;       // 2304
    float* x1    = ws + 4368;       // 2048 (h0)
    float* x2    = ws + 6416;       // 2048 (h1)
    float* gi0   = ws + 8464;       // 4 x 6144  = 24576
    float* gh0   = ws + 33040;      // 24576
    float* gi1   = ws + 57616;      // 24576
    float* gh1   = ws + 82192;      // 24576
    float* lg    = ws + 106768;     // 4 x 100

    const int GB = 256;   // 8 waves -> 128 rows per block

    // 1) stack controls
    controls_kernel<<<1, 256, 0, stream>>>(hidden, Wc, bc, ctrl);

    // 2) s_raw partials = Ws @ h2s   (256 x 4096, K split 8)
    gemv_wmma_kernel<4096, 8><<<dim3(2, 8), GB, 0, stream>>>(Ws, hidden, nullptr, s_raw, SW);

    // 3) x0[0:H] = emb[inp]
    embed_kernel<<<(H + 255) / 256, 256, 0, stream>>>(inp, emb, x0);

    // 4) new_stack + x0[H:H+SW]
    stack_kernel<<<SD, SW, 0, stream>>>(stack, s_raw, bs, ctrl, out_stack, x0);

    // 5) layer-0 GEMV partials (K split 4)
    gemv_wmma_kernel<2048, 4><<<dim3(48, 4), GB, 0, stream>>>(Whh0, hidden, nullptr, gh0, 3 * H);
    gemv_wmma_kernel<2304, 4><<<dim3(48, 4), GB, 0, stream>>>(Wih0, x0,     nullptr, gi0, 3 * H);

    // 6) layer-0 gates -> h0
    gru_gate_kernel<<<(H + 255) / 256, 256, 0, stream>>>(gi0, gh0, bih0, bhh0,
                                                         hidden, out_hidden, x1);

    // 7) layer-1 GEMV partials
    gemv_wmma_kernel<2048, 4><<<dim3(48, 4), GB, 0, stream>>>(Wih1, x1,         nullptr, gi1, 3 * H);
    gemv_wmma_kernel<2048, 4><<<dim3(48, 4), GB, 0, stream>>>(Whh1, hidden + H, nullptr, gh1, 3 * H);

    // 8) layer-1 gates -> h1
    gru_gate_kernel<<<(H + 255) / 256, 256, 0, stream>>>(gi1, gh1, bih1, bhh1,
                                                         hidden + H, out_hidden + H, x2);

    // 9) logits partials + finalize
    gemv_wmma_kernel<2048, 4><<<dim3(1, 4), GB, 0, stream>>>(Wd, x2, bd, lg, VOCAB);
    logits_final_kernel<<<1, 128, 0, stream>>>(lg, out_logits);
}